// CodeGraphNN_31155692765403
// MI455X (gfx1250) — compile-verified
//
#include <hip/hip_runtime.h>

// ---- problem constants (match reference) ----
#define NNODES 40000   // divisible by 64 -> 16-row tiles exact, buffers stay aligned
#define NEDGES 640000
#define F_IN   128
#define HDIM   64
#define COUT   32

typedef float v2f __attribute__((ext_vector_type(2)));
typedef float v8f __attribute__((ext_vector_type(8)));

// Only check on the device pass: host (x86) preprocessor can't see amdgcn builtins,
// even though they are callable inside device functions via aux-target builtin support.
#if defined(__HIP_DEVICE_COMPILE__) && !__has_builtin(__builtin_amdgcn_wmma_f32_16x16x4_f32)
#error "device toolchain lacks __builtin_amdgcn_wmma_f32_16x16x4_f32"
#endif

// ---------------- degree / gcn_norm ----------------
__global__ void deg_init_kernel(float* __restrict__ deg, int n) {
  int i = blockIdx.x * blockDim.x + threadIdx.x;
  if (i < n) deg[i] = 1.0f;  // self-loop contributes 1 to every node's degree
}

__global__ void deg_accum_kernel(const int* __restrict__ dst, float* __restrict__ deg, int e) {
  int i = blockIdx.x * blockDim.x + threadIdx.x;
  if (i < e) atomicAdd(&deg[dst[i]], 1.0f);
}

__global__ void deg_rsqrt_kernel(float* __restrict__ deg, int n) {
  int i = blockIdx.x * blockDim.x + threadIdx.x;
  if (i < n) {
    float d = deg[i];
    deg[i] = (d > 0.0f) ? rsqrtf(d) : 0.0f;  // deg >= 1 here, guard kept for fidelity
  }
}

// agg[i][c] = bias[c] + dinv[i]^2 * h[i][c]   (bias + self-loop folded; runs before scatter)
__global__ void agg_init_kernel(const float* __restrict__ h, const float* __restrict__ bias,
                                const float* __restrict__ dinv, float* __restrict__ agg,
                                int n, int lgD) {
  int idx = blockIdx.x * blockDim.x + threadIdx.x;
  if (idx >= (n << lgD)) return;
  int i = idx >> lgD;
  int c = idx & ((1 << lgD) - 1);
  float w = dinv[i];
  agg[idx] = bias[c] + w * w * h[idx];
}

// agg[dst] += dinv[src]*dinv[dst] * h[src]; one thread = one edge x 4 channels (float4 gather)
__global__ void edge_scatter_kernel(const float* __restrict__ h, const int* __restrict__ src,
                                    const int* __restrict__ dst, const float* __restrict__ dinv,
                                    float* __restrict__ agg, int e, int lgG, int d) {
  int tid = blockIdx.x * blockDim.x + threadIdx.x;
  if (tid >= (e << lgG)) return;
  int ei = tid >> lgG;
  int c  = (tid & ((1 << lgG) - 1)) << 2;
  int s = src[ei], t = dst[ei];
  float w = dinv[s] * dinv[t];
  const float4 hv = *(const float4*)(h + (size_t)s * d + c);
  float* ap = agg + (size_t)t * d + c;
  atomicAdd(ap + 0, w * hv.x);
  atomicAdd(ap + 1, w * hv.y);
  atomicAdd(ap + 2, w * hv.z);
  atomicAdd(ap + 3, w * hv.w);
}

// ---------------- fp32 WMMA GEMM: Out[NNODES x Mo] = (relu?)(A[NNODES x K]) * W[K x Mo] (+bias)
// One wave -> one 16x16 output tile, K/4 steps of v_wmma_f32_16x16x4_f32.
// A 16x4 layout: lane l holds A[l&15][k0 + 2*(l>>4) + {0,1}] in 2 VGPRs (ISA 7.12.2).
// B 4x16 layout mirrors with lane&15 as the column. C/D: VGPR v, lane l -> row v + 8*(l>>4).
template <bool RELU_IN>
__global__ void gemm_wmma_kernel(const float* __restrict__ A, const float* __restrict__ W,
                                 const float* __restrict__ bias, float* __restrict__ Out,
                                 int K, int Mo) {
  const int lane = threadIdx.x & 31;
  const int wave = (blockIdx.x * blockDim.x + threadIdx.x) >> 5;
  const int ntc  = Mo >> 4;
  const int tiles = (NNODES >> 4) * ntc;
  if (wave >= tiles) return;                // wave-uniform: EXEC stays all-ones for WMMA
  const int mt = wave / ntc;
  const int nt = wave - mt * ntc;

  const int half = lane >> 4;               // 0: lanes 0-15, 1: lanes 16-31
  const int l15  = lane & 15;
  const int row  = (mt << 4) + l15;         // A row owned by this lane
  const int col  = (nt << 4) + l15;         // B/C/D column owned by this lane
  const int kk   = half << 1;               // K sub-offset within the 4-slice: 0 or 2

  const float* arow = A + (size_t)row * K;
  v8f c = {};
  for (int k0 = 0; k0 < K; k0 += 4) {
    float a0 = arow[k0 + kk];
    float a1 = arow[k0 + kk + 1];
    if (RELU_IN) { a0 = fmaxf(a0, 0.0f); a1 = fmaxf(a1, 0.0f); }
    v2f a; a.x = a0; a.y = a1;
    v2f b;
    b.x = W[(size_t)(k0 + kk) * Mo + col];
    b.y = W[(size_t)(k0 + kk + 1) * Mo + col];
    c = __builtin_amdgcn_wmma_f32_16x16x4_f32(false, a, false, b, (short)0, c, false, false);
  }
  const float bv = bias ? bias[col] : 0.0f;
  const int rbase = (mt << 4) + (half << 3);
  for (int v = 0; v < 8; ++v)
    Out[(size_t)(rbase + v) * Mo + col] = c[v] + bv;  // 16-lane coalesced column stores
}

static inline int gemm_blocks(int Mo) {
  int tiles = (NNODES / 16) * (Mo / 16);
  return (tiles + 7) / 8;                   // 8 waves (256 threads) per block
}

extern "C" void kernel_launch(void* const* d_in, const int* in_sizes, int n_in,
                              void* d_out, int out_size, void* d_ws, size_t ws_size,
                              hipStream_t stream) {
  (void)in_sizes; (void)n_in; (void)out_size; (void)ws_size;

  const float* x  = (const float*)d_in[0];
  const int*   ei = (const int*)d_in[1];     // [2, E] row-major
  const float* W1 = (const float*)d_in[3];
  const float* b1 = (const float*)d_in[4];
  const float* W2 = (const float*)d_in[5];
  const float* b2 = (const float*)d_in[6];
  const float* W3 = (const float*)d_in[7];
  const float* b3 = (const float*)d_in[8];
  const float* W4 = (const float*)d_in[9];
  const float* b4 = (const float*)d_in[10];
  const float* Wl = (const float*)d_in[11];
  const float* bl = (const float*)d_in[12];
  const int* srcE = ei;                      // edge_index[0]
  const int* dstE = ei + NEDGES;             // edge_index[1]

  // workspace: dinv (N f32, 256B-aligned since N%64==0), then two N x 128 ping-pong buffers
  float* dinv = (float*)d_ws;
  float* bufA = dinv + NNODES;
  float* bufB = bufA + (size_t)NNODES * 128;
  float* out  = (float*)d_out;

  // ---- gcn_norm: degree (with self-loops) -> D^{-1/2}, computed in place ----
  deg_init_kernel <<<(NNODES + 255) / 256, 256, 0, stream>>>(dinv, NNODES);
  deg_accum_kernel<<<(NEDGES + 255) / 256, 256, 0, stream>>>(dstE, dinv, NEDGES);
  deg_rsqrt_kernel<<<(NNODES + 255) / 256, 256, 0, stream>>>(dinv, NNODES);

  // ---- layer 1: x(128) @ W1 -> 64, aggregate; ReLU deferred into next GEMM ----
  gemm_wmma_kernel<false><<<gemm_blocks(64), 256, 0, stream>>>(x, W1, nullptr, bufA, 128, 64);
  agg_init_kernel<<<((NNODES << 6) + 255) / 256, 256, 0, stream>>>(bufA, b1, dinv, bufB, NNODES, 6);
  edge_scatter_kernel<<<((NEDGES << 4) + 255) / 256, 256, 0, stream>>>(bufA, srcE, dstE, dinv, bufB, NEDGES, 4, 64);

  // ---- layer 2: relu(h1)(64) @ W2 -> 128, aggregate ----
  gemm_wmma_kernel<true><<<gemm_blocks(128), 256, 0, stream>>>(bufB, W2, nullptr, bufA, 64, 128);
  agg_init_kernel<<<((NNODES << 7) + 255) / 256, 256, 0, stream>>>(bufA, b2, dinv, bufB, NNODES, 7);
  edge_scatter_kernel<<<((NEDGES << 5) + 255) / 256, 256, 0, stream>>>(bufA, srcE, dstE, dinv, bufB, NEDGES, 5, 128);

  // ---- layer 3: relu(h2)(128) @ W3 -> 128, aggregate ----
  gemm_wmma_kernel<true><<<gemm_blocks(128), 256, 0, stream>>>(bufB, W3, nullptr, bufA, 128, 128);
  agg_init_kernel<<<((NNODES << 7) + 255) / 256, 256, 0, stream>>>(bufA, b3, dinv, bufB, NNODES, 7);
  edge_scatter_kernel<<<((NEDGES << 5) + 255) / 256, 256, 0, stream>>>(bufA, srcE, dstE, dinv, bufB, NEDGES, 5, 128);

  // ---- layer 4: relu(h3)(128) @ W4 -> 64, aggregate ----
  gemm_wmma_kernel<true><<<gemm_blocks(64), 256, 0, stream>>>(bufB, W4, nullptr, bufA, 128, 64);
  agg_init_kernel<<<((NNODES << 6) + 255) / 256, 256, 0, stream>>>(bufA, b4, dinv, bufB, NNODES, 6);
  edge_scatter_kernel<<<((NEDGES << 4) + 255) / 256, 256, 0, stream>>>(bufA, srcE, dstE, dinv, bufB, NEDGES, 4, 64);

  // ---- final linear: relu(h4)(64) @ Wl + bl -> out[N, 32] ----
  gemm_wmma_kernel<true><<<gemm_blocks(32), 256, 0, stream>>>(bufB, Wl, bl, out, 64, 32);
}